// EdgeGCN_1580547965887
// MI455X (gfx1250) — compile-verified
//
#include <hip/hip_runtime.h>

typedef float v2f __attribute__((ext_vector_type(2)));
typedef float v8f __attribute__((ext_vector_type(8)));

#define NN   1536
#define FF   64
#define FIN  65      // 64 feat channels + 1 adjacency channel
#define NOUT 64
#define NCHUNK 17    // 16 K=4 chunks for |feat diff| + 1 chunk for adjacency

// One block per source node i, 8 waves. Each wave owns a full 16x64 output
// row-tile: all four 16-col B tiles (W^T fragments, 17 K=4 chunks each) stay
// resident in VGPRs, so the A matrix (|feat[j]-feat[i]| rows + adj channel)
// is built exactly once per row-tile and feeds 4 independent WMMA chains
// (68 v_wmma_f32_16x16x4_f32 per 4KB of output).
__global__ __launch_bounds__(256, 1) void edge_gcn_wmma(
    const float* __restrict__ feat,   // [N, F]
    const float* __restrict__ adj,    // [N, N]
    const float* __restrict__ W,      // [OUT, FIN] row-major
    const float* __restrict__ bias,   // [OUT]
    float* __restrict__ out)          // [N*N, OUT]
{
    const int i    = blockIdx.x;
    const int tid  = threadIdx.x;
    const int lane = tid & 31;
    const int wave = tid >> 5;      // 0..7 -> 8-way interleave over j-tiles

    const int l15  = lane & 15;
    const int hi   = lane >> 4;     // lane half: k%4 in {0,1} (lo) or {2,3} (hi)
    const int koff = hi * 2;

    // ---- B fragments, persistent: B[k][n] = W[nt*16+n][k], all 4 col-tiles ----
    v2f Bf[4][NCHUNK];
    float bval[4];
    #pragma unroll
    for (int nt = 0; nt < 4; ++nt) {
        const int o = nt * 16 + l15;            // output column this lane serves
        const float* wrow = W + o * FIN;
        #pragma unroll
        for (int t = 0; t < 16; ++t) {
            Bf[nt][t].x = wrow[4 * t + koff];
            Bf[nt][t].y = wrow[4 * t + koff + 1];
        }
        // chunk 16: k=64 (adjacency weight) in the k%4==0 slot, zero padding
        Bf[nt][16].x = hi ? 0.0f : wrow[64];
        Bf[nt][16].y = 0.0f;
        bval[nt] = bias[o];
    }

    // ---- feat[i] fragments, persistent (k = 4t+koff, 4t+koff+1) ----
    float fi0[16], fi1[16];
    {
        const float* fr = feat + i * FF;
        #pragma unroll
        for (int t = 0; t < 16; ++t) {
            fi0[t] = fr[4 * t + koff];
            fi1[t] = fr[4 * t + koff + 1];
        }
    }

    for (int jt = wave; jt < NN / 16; jt += 8) {
        const int j0 = jt * 16;
        const float* fj = feat + (j0 + l15) * FF;   // A-matrix row for this lane

        // prefetch the feat rows this lane needs next iteration (j += 128)
        if (j0 + 128 < NN)
            __builtin_prefetch(fj + 128 * FF, 0, 1);

        // adjacency channel A[m][64]; zero in the upper lane-half (k%4!=0)
        const float av = hi ? 0.0f : adj[(size_t)i * NN + j0 + l15];

        v8f c[4];
        #pragma unroll
        for (int nt = 0; nt < 4; ++nt)
            #pragma unroll
            for (int v = 0; v < 8; ++v)
                c[nt][v] = bval[nt];                 // bias folded into C init

        #pragma unroll
        for (int t = 0; t < 16; ++t) {
            v2f a;
            a.x = __builtin_fabsf(fj[4 * t + koff]     - fi0[t]);
            a.y = __builtin_fabsf(fj[4 * t + koff + 1] - fi1[t]);
            #pragma unroll
            for (int nt = 0; nt < 4; ++nt)
                c[nt] = __builtin_amdgcn_wmma_f32_16x16x4_f32(
                    false, a, false, Bf[nt][t], (short)0, c[nt], false, false);
        }
        {   // adjacency chunk (K = 64..67, 65..67 are zero padding)
            v2f a; a.x = av; a.y = 0.0f;
            #pragma unroll
            for (int nt = 0; nt < 4; ++nt)
                c[nt] = __builtin_amdgcn_wmma_f32_16x16x4_f32(
                    false, a, false, Bf[nt][16], (short)0, c[nt], false, false);
        }

        // ---- store: C tile nt, VGPR v -> row j0 + v + 8*hi, col nt*16 + l15 ----
        float* op = out + ((size_t)i * NN + j0 + 8 * hi) * NOUT + l15;
        #pragma unroll
        for (int nt = 0; nt < 4; ++nt)
            #pragma unroll
            for (int v = 0; v < 8; ++v)
                __builtin_nontemporal_store(c[nt][v], op + v * NOUT + nt * 16);
    }
}

extern "C" void kernel_launch(void* const* d_in, const int* in_sizes, int n_in,
                              void* d_out, int out_size, void* d_ws, size_t ws_size,
                              hipStream_t stream) {
    const float* feat = (const float*)d_in[0];  // [1536, 64]
    const float* adj  = (const float*)d_in[1];  // [1536, 1536]
    const float* W    = (const float*)d_in[2];  // [64, 65]
    const float* bias = (const float*)d_in[3];  // [64]
    float* out        = (float*)d_out;          // [1536*1536, 64]
    (void)in_sizes; (void)n_in; (void)out_size; (void)d_ws; (void)ws_size;

    edge_gcn_wmma<<<dim3(NN), dim3(256), 0, stream>>>(feat, adj, W, bias, out);
}